// GraphAttentionLayer_68470368633044
// MI455X (gfx1250) — compile-verified
//
#include <hip/hip_runtime.h>
#include <hip/hip_bf16.h>

#define NODES 50000
#define EDGES 800000
#define DIMV 96
#define HEADSV 4
#define CH 24
#define NEG_SLOPE 0.2f

typedef __attribute__((ext_vector_type(2))) float v2f;
typedef __attribute__((ext_vector_type(8))) float v8f;

// ---- float <-> order-preserving int mapping (for atomicMax on floats) ----
__device__ __forceinline__ int f2ord(float f) {
    int i = __float_as_int(f);
    return (i >= 0) ? i : (i ^ 0x7fffffff);
}
__device__ __forceinline__ float ord2f(int i) {
    int j = (i >= 0) ? i : (i ^ 0x7fffffff);
    return __int_as_float(j);
}
// f2ord(-inf) = 0xFF800000 ^ 0x7FFFFFFF = 0x807FFFFF
#define ORD_NEG_INF 0x807FFFFF

// ---------------------------------------------------------------------------
// Init: zero the output accumulator, set segment-max to -inf (ordered-int),
// zero softmax denominators.
// ---------------------------------------------------------------------------
__global__ void gat_init(float* __restrict__ xout, int* __restrict__ mord,
                         float* __restrict__ den) {
    int i = blockIdx.x * blockDim.x + threadIdx.x;
    if (i < NODES * DIMV) xout[i] = 0.0f;
    if (i < NODES * HEADSV) { mord[i] = ORD_NEG_INF; den[i] = 0.0f; }
}

// ---------------------------------------------------------------------------
// WMMA GEMM: H = X @ W, fp32, V_WMMA_F32_16X16X4_F32.
// One block = 6 waves = one 16-row tile; wave w computes columns [16w,16w+16).
// X tile (16x96) and full W (96x96) staged through LDS.
// ---------------------------------------------------------------------------
__global__ __launch_bounds__(192) void gat_gemm(const float* __restrict__ X,
                                                const float* __restrict__ W,
                                                float* __restrict__ H) {
    __shared__ float Xs[16 * DIMV];     //  6 KB
    __shared__ float Ws[DIMV * DIMV];   // 36 KB
    const int tid = threadIdx.x;
    const int rt  = blockIdx.x;                    // row tile, 0..3124
    const float* Xt = X + (size_t)rt * 16 * DIMV;  // contiguous 16x96 chunk
    for (int t = tid; t < 16 * DIMV; t += 192) Xs[t] = Xt[t];
    for (int t = tid; t < DIMV * DIMV; t += 192) Ws[t] = W[t];
    __syncthreads();

    const int wave = tid >> 5;     // column tile 0..5
    const int lane = tid & 31;
    const int half = lane >> 4;    // 0: lanes 0-15, 1: lanes 16-31
    const int l16  = lane & 15;

    v8f acc = {};
    #pragma unroll
    for (int k = 0; k < DIMV; k += 4) {
        const int kk = k + half * 2;
        v2f a, b;
        // A 16x4 (MxK): lane row = l16, VGPR0/1 hold K = kk, kk+1
        a.x = Xs[l16 * DIMV + kk];
        a.y = Xs[l16 * DIMV + kk + 1];
        // B 4x16 (KxN): lane col = l16, VGPR0/1 hold rows K = kk, kk+1
        b.x = Ws[kk * DIMV + wave * 16 + l16];
        b.y = Ws[(kk + 1) * DIMV + wave * 16 + l16];
        acc = __builtin_amdgcn_wmma_f32_16x16x4_f32(
            /*neg_a=*/false, a, /*neg_b=*/false, b,
            /*c_mod=*/(short)0, acc, /*reuse_a=*/false, /*reuse_b=*/false);
    }
    // C/D 16x16 f32: VGPR v holds row (v + 8*half), col l16
    #pragma unroll
    for (int v = 0; v < 8; ++v)
        H[(size_t)(rt * 16 + half * 8 + v) * DIMV + wave * 16 + l16] = acc[v];
}

// ---------------------------------------------------------------------------
// Per-(node,head) attention logits: alpha_s/alpha_d = <h[n,h,:], att[h,:]>
// ---------------------------------------------------------------------------
__global__ void gat_alpha(const float* __restrict__ H,
                          const float* __restrict__ att_src,
                          const float* __restrict__ att_dst,
                          float* __restrict__ as_, float* __restrict__ ad_) {
    int i = blockIdx.x * blockDim.x + threadIdx.x;
    if (i >= NODES * HEADSV) return;
    int n = i >> 2, hh = i & 3;
    const float* hp = H + (size_t)n * DIMV + hh * CH;
    const float* s  = att_src + hh * CH;
    const float* d  = att_dst + hh * CH;
    float vs = 0.f, vd = 0.f;
    #pragma unroll
    for (int c = 0; c < CH; ++c) { float hv = hp[c]; vs += hv * s[c]; vd += hv * d[c]; }
    as_[i] = vs;
    ad_[i] = vd;
}

// ---------------------------------------------------------------------------
// Pass 1 over edges: e = leaky_relu(as[src]+ad[dst]); segment max into mord.
// ---------------------------------------------------------------------------
__global__ void gat_edge_max(const int* __restrict__ src, const int* __restrict__ dst,
                             const float* __restrict__ as_, const float* __restrict__ ad_,
                             int* __restrict__ mord) {
    int i = blockIdx.x * blockDim.x + threadIdx.x;
    if (i >= EDGES * HEADSV) return;
    int e = i >> 2, hh = i & 3;
    int s = src[e], d = dst[e];
    float v = as_[s * HEADSV + hh] + ad_[d * HEADSV + hh];
    v = (v > 0.f) ? v : NEG_SLOPE * v;
    atomicMax(&mord[d * HEADSV + hh], f2ord(v));
}

// ---------------------------------------------------------------------------
// Finalize segment max: ordered-int -> float, non-finite (no edges) -> 0.
// Result stored as float bits in the same buffer.
// ---------------------------------------------------------------------------
__global__ void gat_fin_m(int* __restrict__ mord) {
    int i = blockIdx.x * blockDim.x + threadIdx.x;
    if (i >= NODES * HEADSV) return;
    float f = ord2f(mord[i]);
    if ((__float_as_int(f) & 0x7f800000) == 0x7f800000) f = 0.0f;  // inf/nan -> 0
    mord[i] = __float_as_int(f);
}

// ---------------------------------------------------------------------------
// Pass 2 over edges: ex = exp(e - m[dst]); den[dst] += ex; stash ex.
// ---------------------------------------------------------------------------
__global__ void gat_edge_exp(const int* __restrict__ src, const int* __restrict__ dst,
                             const float* __restrict__ as_, const float* __restrict__ ad_,
                             const float* __restrict__ m_, float* __restrict__ ex,
                             float* __restrict__ den) {
    int i = blockIdx.x * blockDim.x + threadIdx.x;
    if (i >= EDGES * HEADSV) return;
    int e = i >> 2, hh = i & 3;
    int s = src[e], d = dst[e];
    float v = as_[s * HEADSV + hh] + ad_[d * HEADSV + hh];
    v = (v > 0.f) ? v : NEG_SLOPE * v;
    float ev = __expf(v - m_[d * HEADSV + hh]);
    ex[i] = ev;
    atomicAdd(&den[d * HEADSV + hh], ev);
}

// ---------------------------------------------------------------------------
// Pass 3 over (edge, dim): xout[dst, dim] += h[src, dim] * alpha(edge, head).
// 96 dims per edge = 3 waves per edge -> src/dst uniform within a wave.
// ---------------------------------------------------------------------------
__global__ void gat_edge_scatter(const int* __restrict__ src, const int* __restrict__ dst,
                                 const float* __restrict__ H, const float* __restrict__ ex,
                                 const float* __restrict__ den, float* __restrict__ xout) {
    long long i = (long long)blockIdx.x * blockDim.x + threadIdx.x;
    if (i >= (long long)EDGES * DIMV) return;
    int e   = (int)(i / DIMV);
    int dim = (int)(i - (long long)e * DIMV);
    int hh  = dim / CH;
    int s = src[e], d = dst[e];
    float alpha = ex[e * HEADSV + hh] / (den[d * HEADSV + hh] + 1e-16f);
    atomicAdd(&xout[(size_t)d * DIMV + dim], H[(size_t)s * DIMV + dim] * alpha);
}

// ---------------------------------------------------------------------------
// Epilogue: x = relu(x + bias), in place.
// ---------------------------------------------------------------------------
__global__ void gat_bias_relu(float* __restrict__ xout, const float* __restrict__ bias) {
    int i = blockIdx.x * blockDim.x + threadIdx.x;
    if (i >= NODES * DIMV) return;
    float v = xout[i] + bias[i % DIMV];
    xout[i] = (v > 0.f) ? v : 0.f;
}

extern "C" void kernel_launch(void* const* d_in, const int* in_sizes, int n_in,
                              void* d_out, int out_size, void* d_ws, size_t ws_size,
                              hipStream_t stream) {
    const float* x    = (const float*)d_in[0];
    const int*   ei   = (const int*)d_in[1];
    const float* W    = (const float*)d_in[2];
    const float* asrc = (const float*)d_in[3];
    const float* adst = (const float*)d_in[4];
    const float* bias = (const float*)d_in[5];
    float* out = (float*)d_out;

    // Workspace layout (floats)
    float* ws  = (float*)d_ws;
    float* xA  = ws;                            // NODES*96
    float* xB  = xA  + (size_t)NODES * DIMV;    // NODES*96
    float* H   = xB  + (size_t)NODES * DIMV;    // NODES*96
    float* as_ = H   + (size_t)NODES * DIMV;    // NODES*4
    float* ad_ = as_ + (size_t)NODES * HEADSV;  // NODES*4
    float* m_  = ad_ + (size_t)NODES * HEADSV;  // NODES*4 (int/float bits)
    float* den = m_  + (size_t)NODES * HEADSV;  // NODES*4
    float* ex  = den + (size_t)NODES * HEADSV;  // EDGES*4

    const int* src = ei;
    const int* dst = ei + EDGES;

    const int blk = 256;
    const int gNode96 = (NODES * DIMV + blk - 1) / blk;
    const int gNodeH  = (NODES * HEADSV + blk - 1) / blk;
    const int gEdgeH  = (EDGES * HEADSV + blk - 1) / blk;
    const long long edims = (long long)EDGES * DIMV;
    const int gEdgeD  = (int)((edims + blk - 1) / blk);

    const float* xin = x;
    for (int l = 0; l < 3; ++l) {
        float* xout = (l == 0) ? xA : (l == 1) ? xB : out;
        gat_init<<<gNode96, blk, 0, stream>>>(xout, (int*)m_, den);
        gat_gemm<<<NODES / 16, 192, 0, stream>>>(xin, W + (size_t)l * DIMV * DIMV, H);
        gat_alpha<<<gNodeH, blk, 0, stream>>>(H, asrc + l * HEADSV * CH,
                                              adst + l * HEADSV * CH, as_, ad_);
        gat_edge_max<<<gEdgeH, blk, 0, stream>>>(src, dst, as_, ad_, (int*)m_);
        gat_fin_m<<<gNodeH, blk, 0, stream>>>((int*)m_);
        gat_edge_exp<<<gEdgeH, blk, 0, stream>>>(src, dst, as_, ad_, m_, ex, den);
        gat_edge_scatter<<<gEdgeD, blk, 0, stream>>>(src, dst, H, ex, den, xout);
        gat_bias_relu<<<gNode96, blk, 0, stream>>>(xout, bias + l * DIMV);
        xin = xout;
    }
}